// SimpleGNN_79508434584074
// MI455X (gfx1250) — compile-verified
//
#include <hip/hip_runtime.h>
#include <math.h>

#define N_NODES   10000
#define N_EDGES   640000
#define D_FEAT    128
#define N_CLASSES 40

typedef __attribute__((ext_vector_type(2))) float v2f;
typedef __attribute__((ext_vector_type(8))) float v8f;

// ---------------------------------------------------------------------------
// Kernel 1: initialize the segment-max accumulator to -inf
// ---------------------------------------------------------------------------
__global__ void sage_init_agg(float* __restrict__ agg) {
    int i = blockIdx.x * blockDim.x + threadIdx.x;
    if (i < N_NODES * D_FEAT) agg[i] = -INFINITY;
}

// Deterministic float atomic max via the classic sign-split integer trick:
// positive floats order like signed ints, negative floats order inversely as
// unsigned ints. Lowers to native global_atomic_max_i32 / min_u32 (L2 atomics).
__device__ __forceinline__ void atomic_max_f32(float* addr, float val) {
    if (!__builtin_signbit(val)) {
        atomicMax((int*)addr, __float_as_int(val));
    } else {
        atomicMin((unsigned int*)addr, __float_as_uint(val));
    }
}

// ---------------------------------------------------------------------------
// Kernel 2: edge scatter-max. One wave per edge; lane handles a float4 slice.
// x (5.1 MB) and agg (5.1 MB) are L2-resident on MI455X (192 MB L2), so this
// is an L2-bandwidth problem: coalesced 512B row gathers + 4 atomics/lane.
// ---------------------------------------------------------------------------
__global__ void sage_scatter_max(const float* __restrict__ x,
                                 const int*   __restrict__ eidx,
                                 float*       __restrict__ agg) {
    unsigned int tid = blockIdx.x * blockDim.x + threadIdx.x;
    unsigned int e   = tid >> 5;          // edge index (one wave-lane-group each)
    int fg           = (tid & 31) * 4;    // feature group: 4 floats per lane
    if (e >= N_EDGES) return;
    int src = eidx[e];
    int dst = eidx[N_EDGES + e];
    const float4 v = *(const float4*)(x + (long long)src * D_FEAT + fg);
    float* base = agg + (long long)dst * D_FEAT + fg;
    atomic_max_f32(base + 0, v.x);
    atomic_max_f32(base + 1, v.y);
    atomic_max_f32(base + 2, v.z);
    atomic_max_f32(base + 3, v.w);
}

// ---------------------------------------------------------------------------
// WMMA panel: accumulate Arow[16 x 128] @ W[128 x 40] into three 16x16 column
// tiles using V_WMMA_F32_16X16X4_F32 (exact fp32, K=4 per step).
// A layout : lanes 0-15 rows, vgpr{0,1} = K {kk, kk+1}, kk = (lane>>4)*2
// B layout : lanes 0-15 cols, vgpr{0,1} = K {kk, kk+1} rows of W
// Tile 2 (cols 32..47) uses a CLAMPED in-bounds index + v_cndmask zeroing so
// the load stays unconditional (no divergent exec-mask save/restore blocks).
// ---------------------------------------------------------------------------
__device__ __forceinline__ void mma_panel(const float* __restrict__ Arow,
                                          const float* __restrict__ W,
                                          int kk, int colb, bool sanitize,
                                          v8f& c0, v8f& c1, v8f& c2) {
    const bool t2ok  = (colb + 32) < N_CLASSES;
    const int  col2c = t2ok ? (colb + 32) : (N_CLASSES - 1);  // clamped, in-bounds
    for (int kb = 0; kb < D_FEAT; kb += 4) {
        float ax = Arow[kb + kk];
        float ay = Arow[kb + kk + 1];
        if (sanitize) {               // PyG: nodes with no in-edges -> 0
            ax = (ax < -1e37f) ? 0.0f : ax;
            ay = (ay < -1e37f) ? 0.0f : ay;
        }
        v2f a; a.x = ax; a.y = ay;
        const float* w0 = W + (kb + kk) * N_CLASSES;
        const float* w1 = w0 + N_CLASSES;
        v2f b0; b0.x = w0[colb];      b0.y = w1[colb];
        v2f b1; b1.x = w0[colb + 16]; b1.y = w1[colb + 16];
        float t2x = w0[col2c];        // unconditional in-bounds load
        float t2y = w1[col2c];
        v2f b2; b2.x = t2ok ? t2x : 0.0f;   // v_cndmask, no exec divergence
                b2.y = t2ok ? t2y : 0.0f;
        c0 = __builtin_amdgcn_wmma_f32_16x16x4_f32(false, a, false, b0,
                                                   (short)0, c0, false, false);
        c1 = __builtin_amdgcn_wmma_f32_16x16x4_f32(false, a, false, b1,
                                                   (short)0, c1, false, false);
        c2 = __builtin_amdgcn_wmma_f32_16x16x4_f32(false, a, false, b2,
                                                   (short)0, c2, false, false);
    }
}

// ---------------------------------------------------------------------------
// Kernel 3: out = log_softmax(agg @ W_l + b_l + x @ W_r).
// One wave per 16 nodes (10000 = 625 * 16, no tail). Both GEMM panels
// accumulate into the same C tiles. Softmax is register-resident using
// __shfl_xor reductions within each 16-lane half (wave32: xor offsets < 16
// never cross halves). EXEC is all-ones at every WMMA.
// ---------------------------------------------------------------------------
__global__ void __launch_bounds__(32)
sage_gemm_logsoftmax(const float* __restrict__ x,
                     const float* __restrict__ agg,
                     const float* __restrict__ Wl,
                     const float* __restrict__ bl,
                     const float* __restrict__ Wr,
                     float*       __restrict__ out) {
    const int lane = threadIdx.x;           // 0..31
    const int m0   = blockIdx.x * 16;       // node tile base
    const int row  = lane & 15;
    const int kk   = (lane >> 4) * 2;
    const int colb = lane & 15;

    v8f c0 = {}, c1 = {}, c2 = {};
    const float* arow = agg + (long long)(m0 + row) * D_FEAT;
    const float* xrow = x   + (long long)(m0 + row) * D_FEAT;

    mma_panel(arow, Wl, kk, colb, true,  c0, c1, c2);  // agg @ W_l
    mma_panel(xrow, Wr, kk, colb, false, c0, c1, c2);  // + x @ W_r

    const bool  t2ok  = (colb + 32) < N_CLASSES;
    const float bias0 = bl[colb];
    const float bias1 = bl[colb + 16];
    const float bias2 = t2ok ? bl[colb + 32] : 0.0f;

    for (int j = 0; j < 8; ++j) {
        const int m = j + (lane >> 4) * 8;             // row in this tile
        float u0 = c0[j] + bias0;
        float u1 = c1[j] + bias1;
        float u2 = t2ok ? (c2[j] + bias2) : -INFINITY;
        // row max across the 40 columns held in this half-wave
        float mx = fmaxf(u0, fmaxf(u1, u2));
        for (int off = 1; off < 16; off <<= 1)
            mx = fmaxf(mx, __shfl_xor(mx, off, 32));
        // log-sum-exp
        float s = expf(u0 - mx) + expf(u1 - mx) + (t2ok ? expf(u2 - mx) : 0.0f);
        for (int off = 1; off < 16; off <<= 1)
            s += __shfl_xor(s, off, 32);
        const float lse = mx + logf(s);
        float* orow = out + (long long)(m0 + m) * N_CLASSES;
        orow[colb]      = u0 - lse;
        orow[colb + 16] = u1 - lse;
        if (t2ok) orow[colb + 32] = u2 - lse;
    }
}

// ---------------------------------------------------------------------------
extern "C" void kernel_launch(void* const* d_in, const int* in_sizes, int n_in,
                              void* d_out, int out_size, void* d_ws, size_t ws_size,
                              hipStream_t stream) {
    const float* x    = (const float*)d_in[0];   // [10000,128]
    const int*   eidx = (const int*)  d_in[1];   // [2,640000]
    const float* Wl   = (const float*)d_in[2];   // [128,40]
    const float* bl   = (const float*)d_in[3];   // [40]
    const float* Wr   = (const float*)d_in[4];   // [128,40]
    float*       out  = (float*)d_out;           // [10000,40]
    float*       agg  = (float*)d_ws;            // [10000,128] scratch

    // 1) agg = -inf
    {
        int total  = N_NODES * D_FEAT;
        int blocks = (total + 255) / 256;
        sage_init_agg<<<blocks, 256, 0, stream>>>(agg);
    }
    // 2) scatter-max over edges: one wave (32 lanes x float4) per edge
    {
        unsigned int threads = (unsigned int)N_EDGES * 32u;   // 20.48M
        unsigned int blocks  = (threads + 255u) / 256u;       // 80000
        sage_scatter_max<<<blocks, 256, 0, stream>>>(x, eidx, agg);
    }
    // 3) fused WMMA GEMMs + bias + log_softmax: 625 waves of 16 nodes each
    {
        sage_gemm_logsoftmax<<<N_NODES / 16, 32, 0, stream>>>(x, agg, Wl, bl, Wr, out);
    }
}